// PagedAttention_34136400068610
// MI455X (gfx1250) — compile-verified
//
#include <hip/hip_runtime.h>
#include <math.h>

typedef __attribute__((ext_vector_type(2))) float v2f;
typedef __attribute__((ext_vector_type(8))) float v8f;

#define NUM_HEADS   32
#define NUM_KV      8
#define QPK         4      // query heads per kv head
#define HEAD_DIM    128
#define BS          16     // tokens per KV block
#define MAXB        256    // max blocks per sequence
#define NWAVES      8      // waves per workgroup (flash-decoding split)
#define ATT_SCALE   0.08838834764831845f

// One workgroup per (seq, kv_head). 8 wave32 waves split the context blocks;
// each wave runs WMMA-based QK^T + online softmax + WMMA-based P*V with its own
// accumulator; waves merge via LDS at the end (log-sum-exp combine).
__global__ __launch_bounds__(NWAVES * 32) void paged_attn_wmma_kernel(
    const float* __restrict__ q,      // [S, 32, 128]
    const float* __restrict__ kc,     // [NB, 8, 16, 16, 8]  (d = hx*8+x)
    const float* __restrict__ vc,     // [NB, 8, 128, 16]
    const int*   __restrict__ bt,     // [S, 256]
    const int*   __restrict__ ctx,    // [S]
    float*       __restrict__ out)    // [S, 32, 128]
{
    __shared__ float p_lds[NWAVES * 64];    // per-wave 4x16 probability tile
    __shared__ float o_lds[NWAVES * 512];   // per-wave partial O (4 heads x 128)
    __shared__ float m_lds[NWAVES * QPK];
    __shared__ float l_lds[NWAVES * QPK];

    const int s     = blockIdx.x >> 3;
    const int g     = blockIdx.x & 7;
    const int tid   = threadIdx.x;
    const int w     = tid >> 5;        // wave id
    const int lane  = tid & 31;
    const int mrow  = lane & 15;       // A-matrix row / B-matrix column index
    const int khalf = lane >> 4;       // selects K={0,1} vs K={2,3} halves

    const int clen = ctx[s];
    const int nblk = (clen + BS - 1) / BS;

    // ---- Q resident as 32 A-fragments (16x4 f32), pre-scaled by 1/sqrt(d) ----
    // A layout (ISA 7.12.2): lane m holds row m; vgpr0 = K=4c+2*khalf, vgpr1 = +1.
    v2f qa[32];
#pragma unroll
    for (int ch = 0; ch < 32; ++ch) {
        v2f v = {0.f, 0.f};
        if (mrow < QPK) {
            const float* p = q + ((size_t)s * NUM_HEADS + g * QPK + mrow) * HEAD_DIM
                               + 4 * ch + 2 * khalf;
            v2f t = *(const v2f*)p;
            v.x = t.x * ATT_SCALE;
            v.y = t.y * ATT_SCALE;
        }
        qa[ch] = v;
    }

    // O accumulator: 8 N-tiles of 16 dims, each a 16x16 f32 C-fragment.
    v8f o[8] = {};
    float mrun[QPK], lrun[QPK];
#pragma unroll
    for (int m = 0; m < QPK; ++m) { mrun[m] = -INFINITY; lrun[m] = 0.f; }

    for (int blk = w; blk < nblk; blk += NWAVES) {
        const int pb = bt[s * MAXB + blk];
        const float* kb = kc + ((size_t)pb * NUM_KV + g) * 2048;  // 16*16*8
        const float* vb = vc + ((size_t)pb * NUM_KV + g) * 2048;  // 128*16

        // prefetch next block for this wave
        if (blk + NWAVES < nblk) {
            const int pn = bt[s * MAXB + blk + NWAVES];
            __builtin_prefetch(kc + ((size_t)pn * NUM_KV + g) * 2048, 0, 1);
            __builtin_prefetch(vc + ((size_t)pn * NUM_KV + g) * 2048, 0, 1);
        }

        // ---- logits tile: C[16 heads(pad) x 16 tokens] = Q * K^T -------------
        // B lane = token column; vgpr pair holds feature rows {d0, d0+1};
        // key cache packs d = (d>>3)*128 + t*8 + (d&7), d0 even -> contiguous pair.
        v8f c = {};
#pragma unroll
        for (int ch = 0; ch < 32; ++ch) {
            const int d0 = 4 * ch + 2 * khalf;
            v2f kk = *(const v2f*)(kb + (d0 >> 3) * (BS * 8) + mrow * 8 + (d0 & 7));
            c = __builtin_amdgcn_wmma_f32_16x16x4_f32(
                    false, qa[ch], false, kk, (short)0, c, false, false);
        }

        // ---- masked online softmax over this 16-token tile (rows 0..3) -------
        const bool valid = (blk * BS + mrow) < clen;
        float pr[QPK], alpha[QPK];
#pragma unroll
        for (int m = 0; m < QPK; ++m) {
            float x = valid ? c[m] : -INFINITY;
            float rmax = x;
            rmax = fmaxf(rmax, __shfl_xor(rmax, 1, 32));
            rmax = fmaxf(rmax, __shfl_xor(rmax, 2, 32));
            rmax = fmaxf(rmax, __shfl_xor(rmax, 4, 32));
            rmax = fmaxf(rmax, __shfl_xor(rmax, 8, 32));
            const float mnew = fmaxf(mrun[m], rmax);
            const float a    = __expf(mrun[m] - mnew);
            const float pv   = valid ? __expf(x - mnew) : 0.f;
            float rsum = pv;
            rsum += __shfl_xor(rsum, 1, 32);
            rsum += __shfl_xor(rsum, 2, 32);
            rsum += __shfl_xor(rsum, 4, 32);
            rsum += __shfl_xor(rsum, 8, 32);
            lrun[m]  = lrun[m] * a + rsum;
            mrun[m]  = mnew;
            pr[m]    = pv;
            alpha[m] = a;
        }

        // rescale accumulator rows 0..3 (rows 4..15 stay exactly zero)
#pragma unroll
        for (int dt = 0; dt < 8; ++dt)
#pragma unroll
            for (int m = 0; m < QPK; ++m)
                o[dt][m] *= alpha[m];

        // ---- relayout P: C-fragment -> A-fragment via per-wave LDS tile ------
        asm volatile("" ::: "memory");
        if (khalf == 0) {
#pragma unroll
            for (int m = 0; m < QPK; ++m)
                p_lds[w * 64 + m * 16 + mrow] = pr[m];
        }
        asm volatile("s_wait_dscnt 0" ::: "memory");
        __builtin_amdgcn_wave_barrier();

        v2f pa[4];
#pragma unroll
        for (int cc = 0; cc < 4; ++cc) {
            v2f t = {0.f, 0.f};
            if (mrow < QPK)
                t = *(const v2f*)&p_lds[w * 64 + mrow * 16 + 4 * cc + 2 * khalf];
            pa[cc] = t;
        }
        asm volatile("" ::: "memory");

        // ---- O += P * V : B lane = dim column, vgpr pair = token rows {t0,t0+1}
        // value cache addr = d*16 + t, t0 even -> contiguous pair.
#pragma unroll
        for (int cc = 0; cc < 4; ++cc) {
            const int t0 = 4 * cc + 2 * khalf;
#pragma unroll
            for (int dt = 0; dt < 8; ++dt) {
                v2f vv = *(const v2f*)(vb + (size_t)(dt * 16 + mrow) * BS + t0);
                o[dt] = __builtin_amdgcn_wmma_f32_16x16x4_f32(
                            false, pa[cc], false, vv, (short)0, o[dt], false, false);
            }
        }
    }

    // ---- publish per-wave partials -----------------------------------------
    if (khalf == 0) {
#pragma unroll
        for (int m = 0; m < QPK; ++m)
#pragma unroll
            for (int dt = 0; dt < 8; ++dt)
                o_lds[w * 512 + m * 128 + dt * 16 + mrow] = o[dt][m];
    }
    if (lane == 0) {
#pragma unroll
        for (int m = 0; m < QPK; ++m) {
            m_lds[w * QPK + m] = mrun[m];
            l_lds[w * QPK + m] = lrun[m];
        }
    }
    __syncthreads();

    // ---- cross-wave log-sum-exp combine: 512 outputs over 256 threads -------
#pragma unroll
    for (int r = 0; r < 2; ++r) {
        const int e = tid + r * 256;
        const int m = e >> 7;
        const int d = e & 127;
        float M = -INFINITY;
#pragma unroll
        for (int ww = 0; ww < NWAVES; ++ww)
            M = fmaxf(M, m_lds[ww * QPK + m]);
        float L = 0.f, acc = 0.f;
#pragma unroll
        for (int ww = 0; ww < NWAVES; ++ww) {
            const float f = __expf(m_lds[ww * QPK + m] - M);
            L   += l_lds[ww * QPK + m] * f;
            acc += o_lds[ww * 512 + m * 128 + d] * f;
        }
        out[((size_t)s * NUM_HEADS + g * QPK + m) * HEAD_DIM + d] = acc / L;
    }
}

extern "C" void kernel_launch(void* const* d_in, const int* in_sizes, int n_in,
                              void* d_out, int out_size, void* d_ws, size_t ws_size,
                              hipStream_t stream) {
    (void)in_sizes; (void)n_in; (void)d_ws; (void)ws_size; (void)out_size;
    const float* q  = (const float*)d_in[0];
    const float* kc = (const float*)d_in[1];
    const float* vc = (const float*)d_in[2];
    const int*   bt = (const int*)d_in[3];
    const int*   cl = (const int*)d_in[4];
    float* out = (float*)d_out;

    dim3 grid(32 * NUM_KV);          // one WG per (seq, kv_head) = 256
    dim3 block(NWAVES * 32);         // 8 wave32 waves
    paged_attn_wmma_kernel<<<grid, block, 0, stream>>>(q, kc, vc, bt, cl, out);
}